// SpearmanRankLoss_36292473651841
// MI455X (gfx1250) — compile-verified
//
#include <hip/hip_runtime.h>

#define N            4096
#define SORT_THREADS 1024
#define GRAM_THREADS 256

typedef __attribute__((ext_vector_type(2))) float v2f;
typedef __attribute__((ext_vector_type(8))) float v8f;

// ---------------------------------------------------------------------------
// Kernel 1: one block per vector.
//  - async global->LDS fill (ASYNCcnt path, global_load_async_to_lds_b128)
//  - bitonic sort (descending) of the 4096 values in LDS
//  - PAVA isotonic regression on z_i = (n-i) - s_sorted[i]  (thread 0, O(n))
//  - soft rank for original position recovered by binary search in the sorted
//    array (values are distinct), r[i] = s_sorted[p] + d[p]
// ---------------------------------------------------------------------------
__global__ __launch_bounds__(SORT_THREADS) void softrank_pava_kernel(
    const float* __restrict__ x0, const float* __restrict__ x1,
    float* __restrict__ r_ws) {
  __shared__ __align__(16) float vals[N];  // sorted values (descending)
  __shared__ float          dual[N];       // d = isotonic_increasing(z)
  __shared__ float          bsum[N];       // PAVA block sums
  __shared__ unsigned short bcnt[N];       // PAVA block counts (<= 4096)

  const float* __restrict__ x = (blockIdx.x == 0) ? x0 : x1;
  float* __restrict__ r = r_ws + (size_t)blockIdx.x * N;
  const int tid = threadIdx.x;

  // ---- async global -> LDS fill: 16 B per thread, 16 KB total --------------
  {
    unsigned lds_off = (unsigned)(size_t)(&vals[0]) + (unsigned)(tid * 16);
    unsigned goff    = (unsigned)(tid * 16);
    asm volatile("global_load_async_to_lds_b128 %0, %1, %2"
                 :: "v"(lds_off), "v"(goff), "s"(x)
                 : "memory");
    asm volatile("s_wait_asynccnt 0x0" ::: "memory");
  }
  __syncthreads();

  // ---- bitonic sort, descending --------------------------------------------
  for (unsigned k = 2; k <= (unsigned)N; k <<= 1) {
    for (unsigned j = k >> 1; j > 0; j >>= 1) {
      for (unsigned i = (unsigned)tid; i < (unsigned)N; i += SORT_THREADS) {
        unsigned ixj = i ^ j;
        if (ixj > i) {
          float a = vals[i], b = vals[ixj];
          bool desc = ((i & k) == 0);  // overall descending order
          if (desc ? (a < b) : (a > b)) { vals[i] = b; vals[ixj] = a; }
        }
      }
      __syncthreads();
    }
  }

  // ---- PAVA (increasing fit) on z_i = (n-i) - vals[i] ----------------------
  if (tid == 0) {
    int nb = 0;
    for (int i = 0; i < N; ++i) {
      float cs = (float)(N - i) - vals[i];
      int   cc = 1;
      // merge while previous block mean >= current block mean
      while (nb > 0 && bsum[nb - 1] * (float)cc >= cs * (float)bcnt[nb - 1]) {
        cs += bsum[nb - 1];
        cc += (int)bcnt[nb - 1];
        --nb;
      }
      bsum[nb] = cs;
      bcnt[nb] = (unsigned short)cc;
      ++nb;
    }
    int pos = 0;
    for (int jb = 0; jb < nb; ++jb) {
      float avg = bsum[jb] / (float)bcnt[jb];
      int   c   = (int)bcnt[jb];
      for (int q = 0; q < c; ++q) dual[pos++] = avg;
    }
  }
  __syncthreads();

  // ---- un-permute: binary search each original value in descending vals ----
  // primal = s_sorted - dual_dec = s_sorted + d  (dual_dec = -d)
  for (int i = tid; i < N; i += SORT_THREADS) {
    float xi = x[i];
    int lo = 0, hi = N - 1;
    while (lo < hi) {
      int mid = (lo + hi) >> 1;
      if (vals[mid] > xi) lo = mid + 1; else hi = mid;
    }
    r[i] = vals[lo] + dual[lo];
  }
}

// ---------------------------------------------------------------------------
// Kernel 2: Gram matrix G = M * M^T with M = [ri; rt; 1; 0...] (16 x 4096),
// accumulated with v_wmma_f32_16x16x4_f32 (8 waves over disjoint K ranges),
// partials combined via ds_add_f32 LDS atomics, then the Pearson loss.
// Fragment construction is branch-free: every lane loads with full EXEC and
// blends via v_cndmask, so the loop body is load -> select -> v_wmma.
// ---------------------------------------------------------------------------
__global__ __launch_bounds__(GRAM_THREADS) void gram_loss_kernel(
    const float* __restrict__ r_ws, float* __restrict__ out) {
  __shared__ float g[9];
  const int tid = threadIdx.x;
  if (tid < 9) g[tid] = 0.0f;
  __syncthreads();

  const int lane  = tid & 31;          // wave32
  const int wave  = tid >> 5;          // 8 waves
  const int m     = lane & 15;         // A-row / B-col handled by this lane
  const int khalf = (lane >> 4) << 1;  // lanes 0-15 -> K {0,1}; 16-31 -> K {2,3}

  const float* __restrict__ ri = r_ws;
  const float* __restrict__ rt = r_ws + N;

  // Loop-invariant per-lane fragment source (rows 2..15 read ri harmlessly;
  // their loaded value is discarded by the select below).
  const float* base   = (m == 1) ? rt : ri;
  const bool   use_ld = (m < 2);
  const float  fill   = (m == 2) ? 1.0f : 0.0f;  // row 2 = ones, 3..15 = zero

  v8f c = {};
  const int kspan = N / 8;             // 512 K-steps per wave
  const float* p = base + wave * kspan + khalf;
  for (int kb = 0; kb < kspan; kb += 4) {
    v2f ld = *(const v2f*)p;           // b64 load, EXEC all-ones
    p += 4;
    v2f a;
    a.x = use_ld ? ld.x : fill;        // v_cndmask
    a.y = use_ld ? ld.y : fill;
    // M rows: 0 = ri, 1 = rt, 2 = ones. A and B fragments are identical per
    // lane (G is symmetric; B = M^T with the mirrored layout). D = A*B + C.
    c = __builtin_amdgcn_wmma_f32_16x16x4_f32(false, a, false, a,
                                              (short)0, c, false, false);
  }

  // C/D layout: VGPR v -> lanes 0-15: (M=v, N=lane); lanes 16-31: (M=v+8).
  // We need G[m][n] for m,n in {0,1,2}: VGPRs 0..2, lanes 0..2.
  if (lane < 3) {
    atomicAdd(&g[0 * 3 + lane], c[0]);
    atomicAdd(&g[1 * 3 + lane], c[1]);
    atomicAdd(&g[2 * 3 + lane], c[2]);
  }
  __syncthreads();

  if (tid == 0) {
    float s_ii = g[0], s_it = g[1], s_i = g[2];
    float s_tt = g[4], s_t = g[5];
    const float nF = (float)N;
    float mi  = s_i / nF, mt = s_t / nF;
    float cov = s_it - nF * mi * mt;
    float vi  = s_ii - nF * mi * mi;
    float vt  = s_tt - nF * mt * mt;
    out[0] = -cov * rsqrtf(vi * vt);
  }
}

extern "C" void kernel_launch(void* const* d_in, const int* in_sizes, int n_in,
                              void* d_out, int out_size, void* d_ws, size_t ws_size,
                              hipStream_t stream) {
  (void)in_sizes; (void)n_in; (void)out_size; (void)ws_size;
  const float* x0 = (const float*)d_in[0];   // input
  const float* x1 = (const float*)d_in[1];   // target
  float* ws  = (float*)d_ws;                 // [0..4095]=ri, [4096..8191]=rt
  float* out = (float*)d_out;

  softrank_pava_kernel<<<2, SORT_THREADS, 0, stream>>>(x0, x1, ws);
  gram_loss_kernel<<<1, GRAM_THREADS, 0, stream>>>(ws, out);
}